// RandomWalk_63024350102322
// MI455X (gfx1250) — compile-verified
//
#include <hip/hip_runtime.h>

#define D_FEAT 64
#define TILE_E 256            // edges per wave (8 groups of 32)
#define WAVES_PER_BLOCK 8

#if defined(__has_builtin)
#  if __has_builtin(__builtin_amdgcn_tensor_load_to_lds) && __has_builtin(__builtin_amdgcn_s_wait_tensorcnt)
#    define USE_TDM 1
#  endif
#endif
#ifndef USE_TDM
#  define USE_TDM 0
#endif

typedef unsigned int u32x4 __attribute__((ext_vector_type(4)));
typedef int          i32x8 __attribute__((ext_vector_type(8)));
typedef int          i32x4 __attribute__((ext_vector_type(4)));

// Zero a buffer with 128-bit stores.
__global__ __launch_bounds__(256) void zero_f4_kernel(float4* __restrict__ p, int n4) {
  int i = blockIdx.x * blockDim.x + threadIdx.x;
  if (i < n4) p[i] = make_float4(0.f, 0.f, 0.f, 0.f);
}

#if USE_TDM
// Byte offset of a generic pointer known to live in LDS.
__device__ __forceinline__ unsigned lds_off(const void* p) {
  return (unsigned)(unsigned long long)
      (__attribute__((address_space(3))) const char*)p;
}

// 1D TDM copy: tile_elems 4-byte elements from gsrc into LDS at lds_byte_off.
// tensor_dim0 = remaining elements so trailing OOB reads return zero (tail tile).
__device__ __forceinline__ void tdm_load_1d(const void* gsrc, unsigned lds_byte_off,
                                            unsigned remaining_elems, unsigned tile_elems) {
  unsigned long long ga = (unsigned long long)(uintptr_t)gsrc;
  u32x4 g0;
  g0.x = 0x1u;                                              // count=1, user descriptor
  g0.y = lds_byte_off;                                      // lds_addr
  g0.z = (unsigned)(ga & 0xFFFFFFFFu);                      // global_addr[31:0]
  g0.w = (unsigned)((ga >> 32) & 0x1FFFFFFu) | (2u << 30);  // global_addr[56:32] | type=2
  i32x8 g1;
  g1[0] = (int)(2u << 16);                                  // data_size=2 (4B); mask/pad/iterate=0
  g1[1] = (int)((remaining_elems & 0xFFFFu) << 16);         // tensor_dim0[15:0] (barrier addr=0)
  g1[2] = (int)((remaining_elems >> 16) | (1u << 16));      // tensor_dim0[31:16] | tensor_dim1=1 lo
  g1[3] = (int)((tile_elems & 0xFFFFu) << 16);              // tensor_dim1 hi=0 | tile_dim0
  g1[4] = 1;                                                // tile_dim1=1, tile_dim2=0 (unused)
  g1[5] = (int)remaining_elems;                             // tensor_dim0_stride[31:0]
  g1[6] = 0;                                                // stride0 hi | stride1 lo
  g1[7] = 0;                                                // stride1 hi
  i32x4 z4 = {0, 0, 0, 0};
  i32x8 z8 = {0, 0, 0, 0, 0, 0, 0, 0};
  // amdgpu-toolchain (clang-23) 6-arg form: (g0, g1, g2, g3, extra_g, cpol)
  __builtin_amdgcn_tensor_load_to_lds(g0, g1, z4, z4, z8, 0);
}
#endif

// COO SpMM scatter: y[row[e], :] += val[e] * x[col[e], :]
// Each wave owns a contiguous TILE_E-edge span; metadata staged via TDM into LDS.
__global__ __launch_bounds__(256) void spmm_coo_scatter_kernel(
    const float* __restrict__ x,
    const float* __restrict__ vals,
    const int*   __restrict__ rows,
    const int*   __restrict__ cols,
    float*       __restrict__ y,
    int n_edges, int n_tiles)
{
  __shared__ float s_vals[WAVES_PER_BLOCK][TILE_E];
  __shared__ int   s_rows[WAVES_PER_BLOCK][TILE_E];
  __shared__ int   s_cols[WAVES_PER_BLOCK][TILE_E];

  const int lane = threadIdx.x & 31;
  const int w    = __builtin_amdgcn_readfirstlane((int)(threadIdx.x >> 5));
  const int tile = blockIdx.x * WAVES_PER_BLOCK + w;
  if (tile >= n_tiles) return;

  const int e0 = tile * TILE_E;
  const unsigned remaining = (unsigned)(n_edges - e0);

#if USE_TDM
  // 3 TDM descriptors replace 24 per-lane vector loads; frees the vmem pipe
  // for gathers + atomics. TENSORcnt orders TDM completion vs LDS reads.
  tdm_load_1d(vals + e0, lds_off(&s_vals[w][0]), remaining, TILE_E);
  tdm_load_1d(rows + e0, lds_off(&s_rows[w][0]), remaining, TILE_E);
  tdm_load_1d(cols + e0, lds_off(&s_cols[w][0]), remaining, TILE_E);
  __builtin_amdgcn_s_wait_tensorcnt(0);
#else
  for (int g = 0; g < TILE_E / 32; ++g) {
    const int idx = g * 32 + lane;
    const int e = e0 + idx;
    if (e < n_edges) {
      s_vals[w][idx] = vals[e];
      s_rows[w][idx] = rows[e];
      s_cols[w][idx] = cols[e];
    }
  }
  // Wave-private LDS region; DS ops are in-order within a wave -> no barrier.
#endif

  const int ngroups = min(TILE_E / 32, (int)((remaining + 31u) >> 5));
  for (int g = 0; g < ngroups; ++g) {
    const int idx = g * 32 + lane;
    const float v = s_vals[w][idx];
    const int   r = s_rows[w][idx];
    const int   c = s_cols[w][idx];
    const int cnt = min(32, n_edges - (e0 + g * 32));
    for (int j = 0; j < cnt; ++j) {
      // Uniform lane index -> scalar readlane broadcast, zero LDS traffic.
      const float vj = __int_as_float(__builtin_amdgcn_readlane(__float_as_int(v), j));
      const int   cj = __builtin_amdgcn_readlane(c, j);
      const int   rj = __builtin_amdgcn_readlane(r, j);

      // 256B coalesced gather of row x[cj] (global_load_b64 per lane).
      const float2 h = *reinterpret_cast<const float2*>(
          x + (size_t)cj * D_FEAT + (lane << 1));

      float* dst = y + (size_t)rj * D_FEAT + (lane << 1);
      // Hardware fp32 atomics, agent scope, no return -> resolved in L2.
      __hip_atomic_fetch_add(dst,     vj * h.x, __ATOMIC_RELAXED, __HIP_MEMORY_SCOPE_AGENT);
      __hip_atomic_fetch_add(dst + 1, vj * h.y, __ATOMIC_RELAXED, __HIP_MEMORY_SCOPE_AGENT);
    }
  }
}

extern "C" void kernel_launch(void* const* d_in, const int* in_sizes, int n_in,
                              void* d_out, int out_size, void* d_ws, size_t ws_size,
                              hipStream_t stream) {
  const float* x    = (const float*)d_in[0];
  const float* vals = (const float*)d_in[1];
  const int*   rows = (const int*)d_in[2];
  const int*   cols = (const int*)d_in[3];
  float* out = (float*)d_out;
  float* b0  = (float*)d_ws;   // one N*D fp32 ping-pong buffer (12.8 MB)

  const int nd = in_sizes[0];             // N * D_FEAT
  const int E  = in_sizes[1];             // number of edges
  const int NUM_STEPS = 5;

  const int n4 = nd / 4;
  const int zeroBlocks = (n4 + 255) / 256;
  const int n_tiles    = (E + TILE_E - 1) / TILE_E;
  const int spmmBlocks = (n_tiles + WAVES_PER_BLOCK - 1) / WAVES_PER_BLOCK;

  // dst alternates so the final step lands in d_out:
  //   s=0: x -> out, s=1: out -> b0, s=2: b0 -> out, s=3: out -> b0, s=4: b0 -> out
  const float* src = x;
  for (int s = 0; s < NUM_STEPS; ++s) {
    float* dst = (((NUM_STEPS - 1 - s) & 1) == 0) ? out : b0;
    zero_f4_kernel<<<zeroBlocks, 256, 0, stream>>>((float4*)dst, n4);
    spmm_coo_scatter_kernel<<<spmmBlocks, 256, 0, stream>>>(src, vals, rows, cols, dst, E, n_tiles);
    src = dst;
  }
}